// RoISampler_74036646249073
// MI455X (gfx1250) — compile-verified
//
#include <hip/hip_runtime.h>
#include <hip/hip_bf16.h>
#include <cstdint>

// Problem constants (from reference)
#define BATCH   32
#define NBOX    8000
#define MGT     256
#define LTOT    (NBOX + MGT)      // 8256
#define KSAMP   512
#define MAXPOS  128
#define NBIN    2048
#define NCHUNK  ((LTOT + 255) / 256)   // 33

typedef unsigned int v4u __attribute__((ext_vector_type(4)));
typedef int          v8i __attribute__((ext_vector_type(8)));
typedef int          v4i __attribute__((ext_vector_type(4)));

// ---------------------------------------------------------------------------
// Tensor Data Mover: DMA `n` contiguous f32 elements (n <= 65535) from global
// memory into LDS (CDNA5 ISA descriptor layout).
// Issued once per wave (EXEC ignored); caller guards to wave 0 and waits on
// TENSORcnt before the block barrier.
// ---------------------------------------------------------------------------
__device__ __forceinline__ void tdm_copy_f32_to_lds(const float* gsrc,
                                                    void* lds_dst,
                                                    unsigned n) {
    unsigned long long ga = (unsigned long long)(uintptr_t)gsrc;
    v4u g0;
    g0.x = 1u;                                         // count = 1 valid descriptor
    g0.y = (unsigned)(uintptr_t)lds_dst;               // LDS byte address
    g0.z = (unsigned)ga;                               // global_addr[31:0]
    g0.w = ((unsigned)(ga >> 32) & 0x01FFFFFFu)        // global_addr[56:32]
           | 0x80000000u;                              // type = 2 at [127:126]
    v8i g1;
    g1[0] = (int)0x00020000u;                          // data_size = 2 -> 4 bytes
    g1[1] = (int)((n & 0xFFFFu) << 16);                // tensor_dim0[15:0]
    g1[2] = (int)(((n >> 16) & 0xFFFFu) | (1u << 16)); // tensor_dim0[31:16], tensor_dim1=1
    g1[3] = (int)((n & 0xFFFFu) << 16);                // tile_dim0 = n
    g1[4] = 0;                                         // tile_dim1/2 = 0 (unused)
    g1[5] = (int)n;                                    // tensor_dim0_stride low 32
    g1[6] = 0;
    g1[7] = 0;
    v4i z4 = {0, 0, 0, 0};                             // groups 2/3 unused (<=2D)
#if defined(__clang_major__) && (__clang_major__ >= 23)
    v8i z8 = {0, 0, 0, 0, 0, 0, 0, 0};
    __builtin_amdgcn_tensor_load_to_lds(g0, g1, z4, z4, z8, 0);
#else
    __builtin_amdgcn_tensor_load_to_lds(g0, g1, z4, z4, 0);
#endif
}

// ---------------------------------------------------------------------------
// Kernel A: matcher. One block = 256 anchors of one batch. gt tile (4 KB)
// arrives in LDS via TDM; IoU max/argmax runs out of LDS broadcasts.
// ---------------------------------------------------------------------------
__global__ __launch_bounds__(256) void roi_match_kernel(
    const float* __restrict__ boxes,     // [B, N, 4]
    const float* __restrict__ gt_boxes,  // [B, M, 4]
    int* __restrict__ ws_mi,             // [B, L] matched gt index
    int* __restrict__ ws_ind)            // [B, L] match indicator
{
    __shared__ float4 s_gt[MGT];

    const int b = blockIdx.x / NCHUNK;
    const int r = blockIdx.x % NCHUNK;
    const int l = r * 256 + threadIdx.x;

    if (threadIdx.x < 32) {   // wave 0 only: one DMA per block
        tdm_copy_f32_to_lds(gt_boxes + (size_t)b * MGT * 4, (void*)&s_gt[0],
                            MGT * 4);
        __builtin_amdgcn_s_wait_tensorcnt(0);
    }
    __syncthreads();

    if (l >= LTOT) return;

    float4 bb;
    if (l < NBOX) bb = ((const float4*)boxes)[(size_t)b * NBOX + l];
    else          bb = ((const float4*)gt_boxes)[(size_t)b * MGT + (l - NBOX)];

    const float ymin1 = bb.x, xmin1 = bb.y, ymax1 = bb.z, xmax1 = bb.w;
    const float a1   = (ymax1 - ymin1) * (xmax1 - xmin1);
    const bool  inv1 = fmaxf(fmaxf(ymin1, xmin1), fmaxf(ymax1, xmax1)) < 0.0f;

    float best = -2.0f;
    int   bm   = 0;
    #pragma unroll 4
    for (int m = 0; m < MGT; ++m) {
        const float4 g = s_gt[m];
        const bool inv2 = fmaxf(fmaxf(g.x, g.y), fmaxf(g.z, g.w)) < 0.0f;
        float ih = fminf(ymax1, g.z) - fmaxf(ymin1, g.x);
        float iw = fminf(xmax1, g.w) - fmaxf(xmin1, g.y);
        ih = fmaxf(ih, 0.0f);
        iw = fmaxf(iw, 0.0f);
        const float inter = ih * iw;
        const float a2    = (g.z - g.x) * (g.w - g.y);
        const float un    = a1 + a2 - inter;
        float sim = (un > 0.0f) ? (inter / un) : 0.0f;
        sim = (inv1 || inv2) ? -1.0f : sim;
        if (sim > best) { best = sim; bm = m; }   // first-max => argmax semantics
    }

    // searchsorted([0,.5,.5], v, 'right') -> INDICATORS[-3,-1,-2,1]
    const int indicator = (best < 0.0f) ? -3 : ((best < 0.5f) ? -1 : 1);

    ws_mi [(size_t)b * LTOT + l] = bm;
    ws_ind[(size_t)b * LTOT + l] = indicator;
}

// ---------------------------------------------------------------------------
// Histogram top-k selection among elements with s_ind[i]==maskval, ranked by
// rand descending with argsort-stable ties (rj > ri || (rj==ri && j<i)).
// Marks selected elements in s_sel; optionally counts them into *s_cnt.
// O(L) + O(boundary_bin^2) instead of O(L^2).
// ---------------------------------------------------------------------------
__device__ __forceinline__ int rand_bin(float r) {
    int bi = (int)(r * (float)NBIN);
    bi = bi < 0 ? 0 : bi;
    return bi > (NBIN - 1) ? (NBIN - 1) : bi;
}

__device__ void topk_select(const float* s_rand, const char* s_ind,
                            char* s_sel, int* s_hist,
                            int* s_t, int* s_above, int* s_cnt,
                            int maskval, int k, int tid, bool count) {
    // 1) histogram of masked candidates
    for (int i = tid; i < NBIN; i += 256) s_hist[i] = 0;
    __syncthreads();
    for (int i = tid; i < LTOT; i += 256)
        if (s_ind[i] == maskval)
            atomicAdd(&s_hist[rand_bin(s_rand[i])], 1);
    __syncthreads();

    // 2) scan from the top bin for threshold bin t and above = suffix(t+1)
    if (tid == 0) {
        int suf = 0, t = -1, above = 0;
        for (int bb = NBIN - 1; bb >= 0; --bb) {
            const int c = s_hist[bb];
            if (suf + c >= k) { t = bb; above = suf; break; }
            suf += c;
        }
        *s_t = t;          // t == -1 -> fewer than k candidates: select all
        *s_above = above;
    }
    __syncthreads();
    const int t = *s_t, above = *s_above;

    // 3) mark selections; exact stable rank only inside the boundary bin
    int local = 0;
    for (int i = tid; i < LTOT; i += 256) {
        if (s_ind[i] != maskval) continue;
        const int bi = rand_bin(s_rand[i]);
        char sel = 0;
        if (t < 0 || bi > t) {
            sel = 1;
        } else if (bi == t) {
            const float ri = s_rand[i];
            int cnt = 0;
            for (int j = 0; j < LTOT; ++j) {
                if (s_ind[j] == maskval) {
                    const float rj = s_rand[j];
                    if (rand_bin(rj) == t)
                        cnt += (rj > ri) || (rj == ri && j < i);
                }
            }
            sel = (above + cnt < k) ? 1 : 0;
        }
        if (sel) { s_sel[i] = 1; local += 1; }
    }
    if (count && local) atomicAdd(s_cnt, local);
    __syncthreads();
}

// ---------------------------------------------------------------------------
// Kernel B: per-batch sampler. One block per batch. rand row (33 KB) lands in
// LDS via TDM. Histogram selection + parallel stable-partition emission.
// ---------------------------------------------------------------------------
__global__ __launch_bounds__(256) void roi_sample_kernel(
    const float* __restrict__ rand,      // [B, L]
    const int*   __restrict__ ws_ind,    // [B, L]
    int* __restrict__ ws_idx)            // [B, K] selected indices
{
    __shared__ float s_rand[LTOT];       // 33024 B (TDM target)
    __shared__ char  s_ind[LTOT];        //  8256 B
    __shared__ char  s_sel[LTOT];        //  8256 B
    __shared__ int   s_hist[NBIN];       //  8192 B
    __shared__ int   s_chunk[NCHUNK + 1];
    __shared__ int   s_t, s_above, s_cnt;

    const int b   = blockIdx.x;
    const int tid = threadIdx.x;

    if (tid < 32)   // wave 0: DMA the rand row
        tdm_copy_f32_to_lds(rand + (size_t)b * LTOT, (void*)&s_rand[0], LTOT);
    for (int j = tid; j < LTOT; j += 256) {
        s_ind[j] = (char)ws_ind[(size_t)b * LTOT + j];
        s_sel[j] = 0;
    }
    if (tid == 0) s_cnt = 0;
    if (tid < 32) __builtin_amdgcn_s_wait_tensorcnt(0);
    __syncthreads();

    // positives: top-128 by rand (stable); counts selected into s_cnt
    topk_select(s_rand, s_ind, s_sel, s_hist, &s_t, &s_above, &s_cnt,
                /*maskval=*/1, /*k=*/MAXPOS, tid, /*count=*/true);

    const int need = KSAMP - s_cnt;      // in [384, 512]

    // negatives: top-(512 - npos_sel) by rand (stable)
    topk_select(s_rand, s_ind, s_sel, s_hist, &s_t, &s_above, &s_cnt,
                /*maskval=*/-1, /*k=*/need, tid, /*count=*/false);

    // ---- top_k(indicator, 512): selected ascending, then unselected ascending
    // parallel stable partition: per-chunk counts -> prefix -> per-chunk emit
    if (tid < NCHUNK) {
        const int lo = tid * 256;
        const int hi = min(LTOT, lo + 256);
        int cnt = 0;
        for (int l = lo; l < hi; ++l) cnt += s_sel[l];
        s_chunk[tid] = cnt;
    }
    __syncthreads();
    if (tid == 0) {
        int acc = 0;
        for (int c = 0; c < NCHUNK; ++c) {
            const int v = s_chunk[c];
            s_chunk[c] = acc;
            acc += v;
        }
        s_chunk[NCHUNK] = acc;           // total selected (<= 512)
    }
    __syncthreads();
    if (tid < NCHUNK) {
        const int lo = tid * 256;
        const int hi = min(LTOT, lo + 256);
        const int total = s_chunk[NCHUNK];
        int k1 = s_chunk[tid];                   // selected write cursor
        int k0 = total + (lo - s_chunk[tid]);    // unselected write cursor
        for (int l = lo; l < hi; ++l) {
            if (s_sel[l]) {
                ws_idx[b * KSAMP + (k1++)] = l;
            } else {
                if (k0 < KSAMP) ws_idx[b * KSAMP + k0] = l;
                k0++;
            }
        }
    }
}

// ---------------------------------------------------------------------------
// Kernel C: gather outputs. Pure bandwidth; b128 stores.
// d_out layout (flat, return order):
//   rois [B,K,4] f32 @0 | gt_boxes [B,K,4] f32 @65536 |
//   classes [B,K] i32 @131072 | indices [B,K] i32 @147456
// ---------------------------------------------------------------------------
__global__ __launch_bounds__(256) void roi_gather_kernel(
    const float* __restrict__ boxes,
    const float* __restrict__ gt_boxes,
    const int*   __restrict__ gt_classes,   // [B, M]
    const int*   __restrict__ ws_mi,
    const int*   __restrict__ ws_ind,
    const int*   __restrict__ ws_idx,
    float* __restrict__ out)
{
    const int t = blockIdx.x * 256 + threadIdx.x;
    if (t >= BATCH * KSAMP) return;
    const int b = t >> 9;          // / 512
    const int l = ws_idx[t];

    float4 bb;
    if (l < NBOX) bb = ((const float4*)boxes)[(size_t)b * NBOX + l];
    else          bb = ((const float4*)gt_boxes)[(size_t)b * MGT + (l - NBOX)];

    const int ind = ws_ind[(size_t)b * LTOT + l];
    const int mi  = ws_mi [(size_t)b * LTOT + l];
    const bool bg = (ind == -1) || (ind == -3);   // negative | invalid

    float4 gb;
    if (bg) { gb.x = gb.y = gb.z = gb.w = 0.0f; }
    else    { gb = ((const float4*)gt_boxes)[(size_t)b * MGT + mi]; }

    float4* o_rois = (float4*)out;
    float4* o_gtb  = (float4*)(out + (size_t)BATCH * KSAMP * 4);
    int*    o_cls  = (int*)  (out + (size_t)BATCH * KSAMP * 8);
    int*    o_idx  = o_cls + (size_t)BATCH * KSAMP;

    o_rois[t] = bb;
    o_gtb[t]  = gb;
    o_cls[t]  = bg ? 0  : gt_classes[(size_t)b * MGT + mi];
    o_idx[t]  = bg ? -1 : mi;
}

// ---------------------------------------------------------------------------
extern "C" void kernel_launch(void* const* d_in, const int* in_sizes, int n_in,
                              void* d_out, int out_size, void* d_ws, size_t ws_size,
                              hipStream_t stream) {
    (void)in_sizes; (void)n_in; (void)out_size; (void)ws_size;

    const float* boxes      = (const float*)d_in[0];   // [32,8000,4]
    const float* gt_boxes   = (const float*)d_in[1];   // [32,256,4]
    const int*   gt_classes = (const int*)  d_in[2];   // [32,256]
    const float* rand       = (const float*)d_in[3];   // [32,8256]

    int* ws_mi  = (int*)d_ws;                    // [B*L]
    int* ws_ind = ws_mi  + (size_t)BATCH * LTOT; // [B*L]
    int* ws_idx = ws_ind + (size_t)BATCH * LTOT; // [B*K]

    roi_match_kernel<<<BATCH * NCHUNK, 256, 0, stream>>>(
        boxes, gt_boxes, ws_mi, ws_ind);

    roi_sample_kernel<<<BATCH, 256, 0, stream>>>(rand, ws_ind, ws_idx);

    roi_gather_kernel<<<(BATCH * KSAMP + 255) / 256, 256, 0, stream>>>(
        boxes, gt_boxes, gt_classes, ws_mi, ws_ind, ws_idx, (float*)d_out);
}